// MLPMambaCore_27943057228440
// MI455X (gfx1250) — compile-verified
//
#include <hip/hip_runtime.h>
#include <hip/hip_bf16.h>
#include <cmath>

typedef __attribute__((ext_vector_type(2))) float v2f;
typedef __attribute__((ext_vector_type(8))) float v8f;

#define BB    2
#define LL    1024
#define DMOD  1024
#define DI    2048
#define NS    16
#define KC    4
#define RR    64
#define XZW   (2 * DI)        // 4096 columns of xz
#define DBLW  (RR + 2 * NS)   // 96 columns of dbl

__device__ __forceinline__ float siluf(float v) {
    return v / (1.0f + __expf(-v));
}
__device__ __forceinline__ float softplusf(float v) {
    return (v > 20.0f) ? v : log1pf(expf(v));
}

// ---------------------------------------------------------------------------
// C[M,N] = A[M,K] (row stride lda) @ W[K,N] (row stride N)
// NT: 16-col sub-tiles per wave (compile-time => branch-free hot loop).
// SP: fused softplus(acc + bias[n]) epilogue.
// One wave computes a 16 x (NT*16) tile via chained V_WMMA_F32_16X16X4_F32,
// reusing the A fragment across the NT column sub-tiles.
// Register double-buffered: k-step s+1's fragments load while step s's
// WMMAs execute (last step peeled => no uniform branch in the loop).
// Block = 128 threads (4 waves) => 64 x (NT*16) block tile.
// Requires: M % 64 == 0, N % (NT*16) == 0, K % 4 == 0, K >= 8.
// ---------------------------------------------------------------------------
template <int NT, bool SP>
__global__ void __launch_bounds__(128) wmma_gemm_f32(
    const float* __restrict__ A, const float* __restrict__ W,
    float* __restrict__ C, int N, int K, int lda,
    const float* __restrict__ bias)
{
    const int wave = threadIdx.x >> 5;
    const int lane = threadIdx.x & 31;
    const int half = lane >> 4;        // 0: lanes 0-15, 1: lanes 16-31
    const int mr   = lane & 15;        // A row (and B/C col) inside 16-tile
    const int m0   = blockIdx.y * 64 + wave * 16;
    const int nb   = blockIdx.x * (NT * 16);

    v8f acc[NT] = {};

    // Incrementing pointers: no 64-bit multiplies in the hot loop.
    const float* aptr = A + (size_t)(m0 + mr) * lda + 2 * half;   // A(m, 2*half)
    const float* wptr = W + (size_t)(2 * half) * N + nb + mr;     // W(2*half, n)

    // Prologue: fragments for k-step 0
    v2f a_cur = *(const v2f*)aptr;
    v2f b_cur[NT];
#pragma unroll
    for (int j = 0; j < NT; ++j) {
        b_cur[j][0] = wptr[j * 16];                // W(ka,   n)
        b_cur[j][1] = wptr[(size_t)N + j * 16];    // W(ka+1, n)
    }

    for (int k0 = 0; k0 < K - 4; k0 += 4) {
        aptr += 4;
        wptr += (size_t)4 * N;
        // Prefetch next k-step's fragments (overlaps with WMMAs below)
        v2f a_nxt = *(const v2f*)aptr;
        v2f b_nxt[NT];
#pragma unroll
        for (int j = 0; j < NT; ++j) {
            b_nxt[j][0] = wptr[j * 16];
            b_nxt[j][1] = wptr[(size_t)N + j * 16];
        }
#pragma unroll
        for (int j = 0; j < NT; ++j)
            acc[j] = __builtin_amdgcn_wmma_f32_16x16x4_f32(
                false, a_cur, false, b_cur[j], (short)0, acc[j], false, false);
        a_cur = a_nxt;
#pragma unroll
        for (int j = 0; j < NT; ++j) b_cur[j] = b_nxt[j];
    }
    // Peeled final k-step
#pragma unroll
    for (int j = 0; j < NT; ++j)
        acc[j] = __builtin_amdgcn_wmma_f32_16x16x4_f32(
            false, a_cur, false, b_cur[j], (short)0, acc[j], false, false);

#pragma unroll
    for (int j = 0; j < NT; ++j) {
        const int col = nb + j * 16 + mr;
        const float bsv = SP ? bias[col] : 0.0f;
#pragma unroll
        for (int r = 0; r < 8; ++r) {
            float v = acc[j][r];
            if (SP) v = softplusf(v + bsv);
            // D layout: vgpr r -> row (r + 8*half), lane&15 -> col
            C[(size_t)(m0 + r + 8 * half) * N + col] = v;
        }
    }
}

// ---------------------------------------------------------------------------
// Causal depthwise conv (K=4) + bias + SiLU over xi = xz[:, :, 0:DI]
// ---------------------------------------------------------------------------
__global__ void conv_silu_kernel(const float* __restrict__ xz,
                                 const float* __restrict__ cw,
                                 const float* __restrict__ cb,
                                 float* __restrict__ u)
{
    const int idx = blockIdx.x * blockDim.x + threadIdx.x;  // over B*L*DI
    const int d  = idx % DI;
    const int bt = idx / DI;       // b*LL + t
    const int t  = bt % LL;
    float acc = cb[d];
    const float* w = cw + d * KC;
#pragma unroll
    for (int k = 0; k < KC; ++k) {
        const int tt = t - (KC - 1) + k;
        if (tt >= 0)
            acc += xz[(size_t)(bt - (KC - 1) + k) * XZW + d] * w[k];
    }
    u[(size_t)bt * DI + d] = siluf(acc);
}

// ---------------------------------------------------------------------------
// Selective scan: one thread per (b, d) channel, h[16] in registers.
// B_t/C_t (32 floats per (b,t)) staged in LDS, shared by the 128 channels
// of this block. Fused skip connection and SiLU gate on z.
// Grid: 32 blocks x 128 threads  (b = blk>>4, d = (blk&15)*128 + tid)
// ---------------------------------------------------------------------------
__global__ void __launch_bounds__(128) scan_kernel(
    const float* __restrict__ dt, const float* __restrict__ u,
    const float* __restrict__ dbl, const float* __restrict__ xz,
    const float* __restrict__ A_log, const float* __restrict__ Dp,
    float* __restrict__ yg)
{
    __shared__ float sBC[2 * NS];
    const int b = blockIdx.x >> 4;
    const int d = (blockIdx.x & 15) * 128 + threadIdx.x;

    float An[NS];
#pragma unroll
    for (int n = 0; n < NS; ++n) An[n] = -__expf(A_log[d * NS + n]);  // A = -exp(A_log)
    const float Dd = Dp[d];
    float h[NS];
#pragma unroll
    for (int n = 0; n < NS; ++n) h[n] = 0.0f;

    for (int t = 0; t < LL; ++t) {
        const size_t row = (size_t)(b * LL + t);
        if (threadIdx.x < 2 * NS)
            sBC[threadIdx.x] = dbl[row * DBLW + RR + threadIdx.x];  // B | C contiguous
        __syncthreads();

        const float dtv = dt[row * DI + d];
        const float uv  = u[row * DI + d];
        const float du  = dtv * uv;
        float y = 0.0f;
#pragma unroll
        for (int n = 0; n < NS; ++n) {
            h[n] = __expf(dtv * An[n]) * h[n] + du * sBC[n];
            y += h[n] * sBC[NS + n];
        }
        const float zv = xz[row * XZW + DI + d];
        yg[row * DI + d] = (y + uv * Dd) * siluf(zv);
        __syncthreads();   // protect sBC before next step's refill
    }
}

// ---------------------------------------------------------------------------
extern "C" void kernel_launch(void* const* d_in, const int* in_sizes, int n_in,
                              void* d_out, int out_size, void* d_ws, size_t ws_size,
                              hipStream_t stream) {
    const float* x          = (const float*)d_in[0];
    const float* in_proj_w  = (const float*)d_in[1];
    const float* conv_w     = (const float*)d_in[2];
    const float* conv_b     = (const float*)d_in[3];
    const float* x_proj_w   = (const float*)d_in[4];
    const float* dt_proj_w  = (const float*)d_in[5];
    const float* dt_proj_b  = (const float*)d_in[6];
    const float* A_log      = (const float*)d_in[7];
    const float* Dvec       = (const float*)d_in[8];
    const float* out_proj_w = (const float*)d_in[9];
    float* out = (float*)d_out;

    // Workspace layout (floats): ~84.7 MB total
    float* ws  = (float*)d_ws;
    float* xz  = ws;                                   // 2048*4096
    float* u   = xz  + (size_t)2048 * 4096;            // 2048*2048
    float* dbl = u   + (size_t)2048 * 2048;            // 2048*96
    float* dtb = dbl + (size_t)2048 * 96;              // 2048*2048
    float* yg  = dtb + (size_t)2048 * 2048;            // 2048*2048

    const dim3 blk(128);
    const int M = BB * LL;                             // 2048 rows everywhere

    // 1) xz = x @ in_proj_w           (2048 x 4096, K=1024)
    wmma_gemm_f32<4, false><<<dim3(4096 / 64, M / 64), blk, 0, stream>>>(
        x, in_proj_w, xz, 4096, DMOD, DMOD, nullptr);

    // 2) u = silu(conv(xi) + b)
    conv_silu_kernel<<<(BB * LL * DI) / 256, 256, 0, stream>>>(xz, conv_w, conv_b, u);

    // 3) dbl = u @ x_proj_w           (2048 x 96, K=2048)  -> 3 blocks of 32 cols
    wmma_gemm_f32<2, false><<<dim3(DBLW / 32, M / 64), blk, 0, stream>>>(
        u, x_proj_w, dbl, DBLW, DI, DI, nullptr);

    // 4) dt = softplus(dbl[:, :64] @ dt_proj_w + dt_proj_b)  (2048 x 2048, K=64, lda=96)
    wmma_gemm_f32<4, true><<<dim3(DI / 64, M / 64), blk, 0, stream>>>(
        dbl, dt_proj_w, dtb, DI, RR, DBLW, dt_proj_b);

    // 5) selective scan + skip + gate -> yg
    scan_kernel<<<32, 128, 0, stream>>>(dtb, u, dbl, xz, A_log, Dvec, yg);

    // 6) out = yg @ out_proj_w        (2048 x 1024, K=2048)
    wmma_gemm_f32<4, false><<<dim3(DMOD / 64, M / 64), blk, 0, stream>>>(
        yg, out_proj_w, out, DMOD, DI, DI, nullptr);
}